// TestNetwork_25718264169060
// MI455X (gfx1250) — compile-verified
//
#include <hip/hip_runtime.h>

// MI455X / gfx1250: fused 12-layer MLP (2 -> 4 x11 -> 1) with
// V_WMMA_F32_16X16X4_F32 (fp32-exact, K=4 matches the net width).
//
// Key layout trick: D.v0 holds row M=0 (lanes 0-15) and row M=8 (lanes 16-31)
// at identical N indices. The next-layer B operand wants feature 0 (lanes
// 0-15, K=0 slot) and feature 2 (lanes 16-31, K=2 slot) at the same N.
// So we replicate the 4 weight rows into A rows {0,1,8,9}; then D.v0/D.v1
// ARE the next B operand after bias+relu: no shuffles, no selects, no
// garbage lanes. Per hidden layer per tile: 1 WMMA + 2 v_add + 2 v_max.
//
// Input: both lane halves load the same 16 rows (addresses coalesce to one
// 128B line), seeding two independent 16-row chains per wave (the
// independent back-to-back WMMAs also cover the WMMA->WMMA RAW hazard).
// Output: Wout replicated into A rows 0 AND 8, so tile1's result lands in
// lanes 16-31 directly -> final merge is a single v_cndmask.

typedef float v2f __attribute__((ext_vector_type(2)));
typedef float v8f __attribute__((ext_vector_type(8)));

#define NLAYER 12

__global__ __launch_bounds__(256) void mlp12_wmma_f32_kernel(
    const float* __restrict__ x,     // [B,2]
    const float* __restrict__ W0,    // [4,2]
    const float* __restrict__ b0,    // [4]
    const float* __restrict__ Wh,    // [10,4,4]
    const float* __restrict__ bh,    // [10,4]
    const float* __restrict__ Wout,  // [1,4]
    const float* __restrict__ bout,  // [1]
    float* __restrict__ out,         // [B,1]
    int nrows)
{
    const int  lane = threadIdx.x & 31;
    const bool lo   = lane < 16;     // lanes 0-15: K=0,1 slots ; 16-31: K=2,3
    const int  m    = lane & 15;     // A-operand M row index for this lane

    // A rows {0,1,8,9} carry weight rows {0,1,2,3}; everything else is zero.
    const int g = (m == 0) ? 0 : (m == 1) ? 1 : (m == 8) ? 2 : (m == 9) ? 3 : -1;

    // ---- A operands (weights) + per-lane-half bias, hoisted out of the loop.
    float wA0[NLAYER], wA1[NLAYER];
    float bB0[NLAYER - 1], bB1[NLAYER - 1];

    // layer 0: W0 is [4,2] -> K=2,3 slots (hi lanes) stay zero
    wA0[0] = (g >= 0 && lo) ? W0[g * 2 + 0] : 0.f;
    wA1[0] = (g >= 0 && lo) ? W0[g * 2 + 1] : 0.f;
    bB0[0] = lo ? b0[0] : b0[2];
    bB1[0] = lo ? b0[1] : b0[3];

#pragma unroll
    for (int i = 0; i < 10; ++i) {
        wA0[i + 1] = (g >= 0) ? Wh[i * 16 + g * 4 + (lo ? 0 : 2)] : 0.f;
        wA1[i + 1] = (g >= 0) ? Wh[i * 16 + g * 4 + (lo ? 1 : 3)] : 0.f;
        bB0[i + 1] = lo ? bh[i * 4 + 0] : bh[i * 4 + 2];
        bB1[i + 1] = lo ? bh[i * 4 + 1] : bh[i * 4 + 3];
    }

    // output layer: Wout in A rows 0 AND 8 -> result lands in both lane halves
    const bool om = (m == 0) || (m == 8);
    wA0[11] = om ? Wout[lo ? 0 : 2] : 0.f;
    wA1[11] = om ? Wout[lo ? 1 : 3] : 0.f;
    const float bo = bout[0];

    const v8f czero = {};  // C = inline 0; bias fused with relu on the VALU

    const int wave   = (int)((blockIdx.x * blockDim.x + threadIdx.x) >> 5);
    const int nwaves = (int)((gridDim.x * blockDim.x) >> 5);
    const int nchunk = nrows >> 5;   // 32 rows per chunk (B = 8.4M divides)

    for (int c = wave; c < nchunk; c += nwaves) {
        const int r0 = c << 5;

        // Both lane halves read the SAME 16 rows: one 128B line per load,
        // non-temporal (stream-once data, keep the 192MB L2 clean).
        const v2f xv0 = __builtin_nontemporal_load((const v2f*)(x + 2 * (r0 + m)));
        const v2f xv1 = __builtin_nontemporal_load((const v2f*)(x + 2 * (r0 + 16 + m)));

        float B0[2], B1[2], o[2];
        B0[0] = lo ? xv0.x : 0.f;  B1[0] = lo ? xv0.y : 0.f;  // K=2,3 -> 0
        B0[1] = lo ? xv1.x : 0.f;  B1[1] = lo ? xv1.y : 0.f;

#pragma unroll
        for (int L = 0; L < NLAYER; ++L) {
#pragma unroll
            for (int t = 0; t < 2; ++t) {
                v2f a; a.x = wA0[L]; a.y = wA1[L];
                v2f b; b.x = B0[t];  b.y = B1[t];
                // D(16x16) = A(16x4) x B(4x16) + 0  [fp32 matrix pipe]
                v8f d = __builtin_amdgcn_wmma_f32_16x16x4_f32(
                            false, a, false, b, (short)0, czero, false, false);
                if (L < NLAYER - 1) {
                    // D.v0/D.v1 are already in next-layer B layout.
                    B0[t] = fmaxf(d[0] + bB0[L], 0.f);
                    B1[t] = fmaxf(d[1] + bB1[L], 0.f);
                } else {
                    o[t] = d[0];  // tile0: lanes 0-15 ; tile1: lanes 16-31
                }
            }
        }

        // single coalesced 4B/lane store, no cross-lane ops
        const float ov = (lo ? o[0] : o[1]) + bo;
        __builtin_nontemporal_store(ov, out + r0 + lane);
    }
}

extern "C" void kernel_launch(void* const* d_in, const int* in_sizes, int n_in,
                              void* d_out, int out_size, void* d_ws, size_t ws_size,
                              hipStream_t stream) {
    const float* x    = (const float*)d_in[0];
    const float* W0   = (const float*)d_in[1];
    const float* b0   = (const float*)d_in[2];
    const float* Wh   = (const float*)d_in[3];
    const float* bh   = (const float*)d_in[4];
    const float* Wout = (const float*)d_in[5];
    const float* bout = (const float*)d_in[6];
    float* out = (float*)d_out;

    const int nrows = in_sizes[0] / 2;  // x is [B,2]

    // ~100 MB of traffic -> memory-bound target ~4.3us @ 23.3 TB/s.
    // 4096 blocks x 8 wave32 = 32768 waves, 8 chunks (256 rows) per wave.
    const int block = 256;
    const int grid  = 4096;
    mlp12_wmma_f32_kernel<<<grid, block, 0, stream>>>(x, W0, b0, Wh, bh,
                                                      Wout, bout, out, nrows);
}